// RelationalAttentionLayer_49237505081935
// MI455X (gfx1250) — compile-verified
//
#include <hip/hip_runtime.h>
#include <stdint.h>

// Problem constants (match reference)
#define Nn   50000
#define Dd   128
#define Hh   4
#define Rr   3
#define ERe  50000
#define HD   512            // H*D
#define EPSv 1e-10f
#define INV_SQRT_HD 0.04419417382415922f   // 1/sqrt(512)

typedef __attribute__((ext_vector_type(2))) float v2f;
typedef __attribute__((ext_vector_type(8))) float v8f;

// ---------------------------------------------------------------------------
// Tiled f32 GEMM using V_WMMA_F32_16X16X4_F32 + async global->LDS A staging.
// C[M,Nc] = A[M,K] @ B[K,Nc].  Requires K%128==0, Nc%64==0. M arbitrary.
// Block = 128 threads = 4 waves. Block tile: 32 rows x 256 cols.
// Wave w: 32 rows x 64 cols at cols [(blockIdx.x*4+w)*64, +64)
//   -> 2 row-subtiles x 4 col-subtiles = 8 WMMA accumulators.
// A tile (32 x 128 per K-chunk) is staged into LDS with
// GLOBAL_LOAD_ASYNC_TO_LDS_B128 and shared by all 4 waves.
// ---------------------------------------------------------------------------
__global__ __launch_bounds__(128) void gemm_wmma_f32(
    const float* __restrict__ A, const float* __restrict__ B,
    float* __restrict__ C, int M, int K, int Nc)
{
    __shared__ float Atile[32 * 132];    // 132-float row stride: bank spread

    const int lane = threadIdx.x & 31;
    const int wave = threadIdx.x >> 5;
    const int col0 = (blockIdx.x * 4 + wave) * 64;
    const int row0 = blockIdx.y * 32;
    const int half = lane >> 4;          // 0: lanes 0-15, 1: lanes 16-31
    const int l16  = lane & 15;
    const bool active = (col0 < Nc);     // wave-uniform

    v8f acc[2][4] = {};

    for (int kc = 0; kc < K; kc += 128) {
        // ---- async-stage A[row0:+32, kc:+128] into LDS (16 KB) ----
        // 32 rows x 32 16B-chunks = 1024 chunks, 8 per thread.
        for (int c = threadIdx.x; c < 1024; c += 128) {
            const int rrow = c >> 5;            // 0..31
            const int ci   = (c & 31) * 4;      // float offset within row
            int gr = row0 + rrow;
            if (gr >= M) gr = M - 1;            // clamp tail rows (stores guarded)
            const float* gsrc = A + (size_t)gr * K + kc + ci;
            const unsigned ldsOff =
                (unsigned)(uintptr_t)(&Atile[rrow * 132 + ci]);
            asm volatile("global_load_async_to_lds_b128 %0, %1, off"
                         :: "v"(ldsOff), "v"(gsrc) : "memory");
        }
        asm volatile("s_wait_asynccnt 0x0" ::: "memory");
        __syncthreads();

        if (active) {
            const float* Al0 = &Atile[l16 * 132];
            const float* Al1 = &Atile[(l16 + 16) * 132];
#pragma unroll 4
            for (int kk = 0; kk < 128; kk += 4) {
                // A fragments from LDS (16x4 each; K pair selected by lane half)
                v2f a0, a1;
                a0[0] = Al0[kk + 2 * half]; a0[1] = Al0[kk + 2 * half + 1];
                a1[0] = Al1[kk + 2 * half]; a1[1] = Al1[kk + 2 * half + 1];

                // B fragments straight from global (L2-resident weights)
                const float* Bk0 = B + (size_t)(kc + kk + 2 * half) * Nc + col0 + l16;
                const float* Bk1 = Bk0 + Nc;
                v2f b0; b0[0] = Bk0[0];  b0[1] = Bk1[0];
                v2f b1; b1[0] = Bk0[16]; b1[1] = Bk1[16];
                v2f b2; b2[0] = Bk0[32]; b2[1] = Bk1[32];
                v2f b3; b3[0] = Bk0[48]; b3[1] = Bk1[48];

                acc[0][0] = __builtin_amdgcn_wmma_f32_16x16x4_f32(false, a0, false, b0, (short)0, acc[0][0], false, false);
                acc[0][1] = __builtin_amdgcn_wmma_f32_16x16x4_f32(false, a0, false, b1, (short)0, acc[0][1], false, false);
                acc[0][2] = __builtin_amdgcn_wmma_f32_16x16x4_f32(false, a0, false, b2, (short)0, acc[0][2], false, false);
                acc[0][3] = __builtin_amdgcn_wmma_f32_16x16x4_f32(false, a0, false, b3, (short)0, acc[0][3], false, false);
                acc[1][0] = __builtin_amdgcn_wmma_f32_16x16x4_f32(false, a1, false, b0, (short)0, acc[1][0], false, false);
                acc[1][1] = __builtin_amdgcn_wmma_f32_16x16x4_f32(false, a1, false, b1, (short)0, acc[1][1], false, false);
                acc[1][2] = __builtin_amdgcn_wmma_f32_16x16x4_f32(false, a1, false, b2, (short)0, acc[1][2], false, false);
                acc[1][3] = __builtin_amdgcn_wmma_f32_16x16x4_f32(false, a1, false, b3, (short)0, acc[1][3], false, false);
            }
        }
        __syncthreads();
    }

    if (active) {
        // C/D layout: VGPR j holds row = base + 8*half + j, col = col0+16*t+l16
#pragma unroll
        for (int rs = 0; rs < 2; ++rs) {
#pragma unroll
            for (int j = 0; j < 8; ++j) {
                const int m = row0 + rs * 16 + 8 * half + j;
                if (m < M) {
                    float* Cp = C + (size_t)m * Nc + col0 + l16;
                    Cp[0]  = acc[rs][0][j];
                    Cp[16] = acc[rs][1][j];
                    Cp[32] = acc[rs][2][j];
                    Cp[48] = acc[rs][3][j];
                }
            }
        }
    }
}

// ---------------------------------------------------------------------------
// Edge scores for one relation: wave per edge.
// lane covers cols [lane*16, +16) of the 512-wide row; head = lane>>3.
// numer[e][h] = relu(score/sqrt(512))^2 + eps ; denom[dst][h] += numer.
// ---------------------------------------------------------------------------
__global__ __launch_bounds__(256) void edge_scores_kernel(
    const float* __restrict__ q, const float* __restrict__ Kr,
    const int* __restrict__ src, const int* __restrict__ dst,
    float* __restrict__ numer, float* __restrict__ denom, int er)
{
    const int lane = threadIdx.x & 31;
    const int e = blockIdx.x * 8 + (threadIdx.x >> 5);
    if (e >= er) return;
    const int s = src[e];
    const int d = dst[e];

    const float4* qp = (const float4*)(q  + (size_t)d * HD + lane * 16);
    const float4* kp = (const float4*)(Kr + (size_t)s * HD + lane * 16);
    float sum = 0.f;
#pragma unroll
    for (int i = 0; i < 4; ++i) {
        const float4 qv = qp[i];
        const float4 kv = kp[i];
        sum += qv.x * kv.x + qv.y * kv.y + qv.z * kv.z + qv.w * kv.w;
    }
    // reduce across the 8 lanes of each head segment
    sum += __shfl_xor(sum, 1, 32);
    sum += __shfl_xor(sum, 2, 32);
    sum += __shfl_xor(sum, 4, 32);

    if ((lane & 7) == 0) {
        const int h = lane >> 3;
        const float sc = sum * INV_SQRT_HD;
        const float rl = sc > 0.f ? sc : 0.f;
        const float nm = rl * rl + EPSv;
        numer[(size_t)e * Hh + h] = nm;
        atomicAdd(&denom[(size_t)d * Hh + h], nm);
    }
}

// ---------------------------------------------------------------------------
// Scatter weighted V into z for one relation: wave per edge.
// ---------------------------------------------------------------------------
__global__ __launch_bounds__(256) void edge_scatter_kernel(
    const float* __restrict__ Vr,
    const int* __restrict__ src, const int* __restrict__ dst,
    const float* __restrict__ numer, const float* __restrict__ denom,
    float* __restrict__ z, int er)
{
    const int lane = threadIdx.x & 31;
    const int e = blockIdx.x * 8 + (threadIdx.x >> 5);
    if (e >= er) return;
    const int s = src[e];
    const int d = dst[e];
    const int h = lane >> 3;

    const float nh = numer[(size_t)e * Hh + h] / denom[(size_t)d * Hh + h];

    const float4* vp = (const float4*)(Vr + (size_t)s * HD + lane * 16);
    float*        zp = z + (size_t)d * HD + lane * 16;
#pragma unroll
    for (int i = 0; i < 4; ++i) {
        const float4 vv = vp[i];
        atomicAdd(&zp[4 * i + 0], nh * vv.x);
        atomicAdd(&zp[4 * i + 1], nh * vv.y);
        atomicAdd(&zp[4 * i + 2], nh * vv.z);
        atomicAdd(&zp[4 * i + 3], nh * vv.w);
    }
}

__global__ void zero_kernel(float* __restrict__ p, size_t n)
{
    const size_t i = (size_t)blockIdx.x * blockDim.x + threadIdx.x;
    if (i < n) p[i] = 0.f;
}

// ---------------------------------------------------------------------------
// Orchestration
// ---------------------------------------------------------------------------
extern "C" void kernel_launch(void* const* d_in, const int* in_sizes, int n_in,
                              void* d_out, int out_size, void* d_ws, size_t ws_size,
                              hipStream_t stream)
{
    (void)in_sizes; (void)n_in; (void)out_size; (void)ws_size;

    const float* x  = (const float*)d_in[0];   // [N, 128]
    const float* WQ = (const float*)d_in[1];   // [128, 512]
    const float* WK = (const float*)d_in[2];   // [3, 128, 512]
    const float* WV = (const float*)d_in[3];   // [3, 128, 512]
    const float* WO = (const float*)d_in[4];   // [512, 128]
    const int*   src = (const int*)d_in[5];    // [3, 50000]
    const int*   dst = (const int*)d_in[6];    // [3, 50000]
    float* out = (float*)d_out;                // [N, 128]

    // Workspace layout (floats), ~310 MB total:
    //   q     : N*512
    //   kv    : N*512 (reused for K_r then V_r)
    //   z     : N*512
    //   denom : N*4
    //   numer : R*ER*4
    float* q     = (float*)d_ws;
    float* kv    = q  + (size_t)Nn * HD;
    float* z     = kv + (size_t)Nn * HD;
    float* denom = z  + (size_t)Nn * HD;
    float* numer = denom + (size_t)Nn * Hh;

    const int rowBlocks = (Nn + 31) / 32;      // 1563 (16-row tail guarded)
    const dim3 blkG(128);
    const dim3 grdG512(2, rowBlocks);          // Nc=512: 8 col tiles / 4 waves
    const dim3 grdG128(1, rowBlocks);          // Nc=128: 2 col tiles
    const dim3 blkE(256);
    const dim3 grdE((ERe + 7) / 8);

    // z and denom are contiguous -> zero both every call (ws poisoned once).
    {
        const size_t nz = (size_t)Nn * HD + (size_t)Nn * Hh;
        zero_kernel<<<dim3((unsigned)((nz + 255) / 256)), dim3(256), 0, stream>>>(z, nz);
    }

    // q = x @ WQ
    gemm_wmma_f32<<<grdG512, blkG, 0, stream>>>(x, WQ, q, Nn, Dd, HD);

    // Pass 1: per relation, K_r = x @ WK[r]; edge scores + denom accumulation
    for (int r = 0; r < Rr; ++r) {
        gemm_wmma_f32<<<grdG512, blkG, 0, stream>>>(x, WK + (size_t)r * Dd * HD, kv, Nn, Dd, HD);
        edge_scores_kernel<<<grdE, blkE, 0, stream>>>(q, kv,
            src + (size_t)r * ERe, dst + (size_t)r * ERe,
            numer + (size_t)r * ERe * Hh, denom, ERe);
    }

    // Pass 2: per relation, V_r = x @ WV[r]; normalized scatter into z
    for (int r = 0; r < Rr; ++r) {
        gemm_wmma_f32<<<grdG512, blkG, 0, stream>>>(x, WV + (size_t)r * Dd * HD, kv, Nn, Dd, HD);
        edge_scatter_kernel<<<grdE, blkE, 0, stream>>>(kv,
            src + (size_t)r * ERe, dst + (size_t)r * ERe,
            numer + (size_t)r * ERe * Hh, denom, z, ERe);
    }

    // out = z @ WO
    gemm_wmma_f32<<<grdG128, blkG, 0, stream>>>(z, WO, out, Nn, HD, Dd);
}